// PureExemplarSoftmaxLoss_17102559773293
// MI455X (gfx1250) — compile-verified
//
#include <hip/hip_runtime.h>
#include <hip/hip_bf16.h>
#include <math.h>

#define SCALE_F   30.0f
#define LAMBDA_F  0.1f
#define EPS_F     1e-6f
#define L2E       1.44269504088896340736f

typedef __attribute__((ext_vector_type(2))) float v2f;
typedef __attribute__((ext_vector_type(4))) float v4f;
typedef __attribute__((ext_vector_type(8))) float v8f;

static __host__ __device__ inline int round_up64(int x) { return (x + 63) & ~63; }

// ---------------------------------------------------------------------------
// Kernel 0: zero the flag and the zero-pad tails of both partial arrays so the
// WMMA reduction can use unconditional 64-value chunks.
// ---------------------------------------------------------------------------
__global__ __launch_bounds__(128)
void init_ws_kernel(float* row_loss, int nrows, int nrows_pad,
                    float* hinge, int nh, int nh_pad, int* flag) {
    const int tid = threadIdx.x;
    if (tid == 0) *flag = 0;
    for (int i = nrows + tid; i < nrows_pad; i += 128) row_loss[i] = 0.0f;
    for (int i = nh + tid;    i < nh_pad;    i += 128) hinge[i]    = 0.0f;
}

// ---------------------------------------------------------------------------
// Kernel 1: per-row online logsumexp over outputs [2B x C]; writes
// row_loss[row] = logsumexp(SCALE*out[row,:]) - SCALE*out[row, label[row]]
// One 256-thread block per row. outputs (328MB) cannot fit the 192MB L2, so
// stream it with non-temporal loads to avoid evicting reusable data.
// ---------------------------------------------------------------------------
__global__ __launch_bounds__(256)
void softmax_rows_kernel(const float* __restrict__ outputs,
                         const int* __restrict__ labels_anchor,
                         const int* __restrict__ labels_neg,
                         float* __restrict__ row_loss,
                         int B, int C) {
    const int row = blockIdx.x;                 // 0 .. 2B-1
    const int tid = threadIdx.x;
    const float* __restrict__ p = outputs + (size_t)row * (size_t)C;

    float m = -__builtin_inff();
    float s = 0.0f;

    const int C4 = C >> 2;                      // float4 count
    const v4f* __restrict__ p4 = (const v4f*)p;
    for (int i = tid; i < C4; i += 256) {
        v4f v = __builtin_nontemporal_load(&p4[i]);   // NT: stream, don't cache
        float xs[4] = {v.x, v.y, v.z, v.w};
#pragma unroll
        for (int k = 0; k < 4; ++k) {
            float t = SCALE_F * xs[k];
            if (t > m) {
                s = s * exp2f((m - t) * L2E) + 1.0f;
                m = t;
            } else {
                s += exp2f((t - m) * L2E);
            }
        }
    }
    // scalar tail (C % 4)
    for (int i = (C4 << 2) + tid; i < C; i += 256) {
        float t = SCALE_F * __builtin_nontemporal_load(&p[i]);
        if (t > m) { s = s * exp2f((m - t) * L2E) + 1.0f; m = t; }
        else       { s += exp2f((t - m) * L2E); }
    }

    // wave32 reduction of (m, s)
#pragma unroll
    for (int off = 16; off >= 1; off >>= 1) {
        float m2 = __shfl_xor(m, off, 32);
        float s2 = __shfl_xor(s, off, 32);
        float mm = fmaxf(m, m2);
        s = s * exp2f((m - mm) * L2E) + s2 * exp2f((m2 - mm) * L2E);
        m = mm;
    }

    __shared__ float sm[8];
    __shared__ float ss[8];
    const int wave = tid >> 5, lane = tid & 31;
    if (lane == 0) { sm[wave] = m; ss[wave] = s; }
    __syncthreads();

    if (tid == 0) {
        float M = sm[0], S = ss[0];
#pragma unroll
        for (int w = 1; w < 8; ++w) {
            float mm = fmaxf(M, sm[w]);
            S = S * exp2f((M - mm) * L2E) + ss[w] * exp2f((sm[w] - mm) * L2E);
            M = mm;
        }
        float lse = M + logf(S);
        int lab = (row < B) ? labels_anchor[row] : labels_neg[row - B];
        row_loss[row] = lse - SCALE_F * p[lab];
    }
}

// ---------------------------------------------------------------------------
// Kernel 2: center term. One wave32 per batch element b:
//   hinge[b] = relu(L1(a,ex_a) - L1(n,ex_a)) + relu(L1(n,ex_n) - L1(a,ex_n))
// Gathered exemplar rows / anchor / negative use default (RT) caching: this
// working set (~74MB) fits the 192MB L2 and is re-read every graph replay.
// ---------------------------------------------------------------------------
__global__ __launch_bounds__(256)
void center_kernel(const float* __restrict__ anchor,
                   const float* __restrict__ negative,
                   const float* __restrict__ exemplars,
                   const int* __restrict__ labels_anchor,
                   const int* __restrict__ labels_neg,
                   float* __restrict__ hinge,
                   int B, int D) {
    const int gwave = (int)((blockIdx.x * 256u + threadIdx.x) >> 5);
    const int lane  = threadIdx.x & 31;
    if (gwave >= B) return;

    const int la = labels_anchor[gwave];
    const int ln = labels_neg[gwave];
    const v4f* __restrict__ pa = (const v4f*)(anchor    + (size_t)gwave * (size_t)D);
    const v4f* __restrict__ pn = (const v4f*)(negative  + (size_t)gwave * (size_t)D);
    const v4f* __restrict__ ea = (const v4f*)(exemplars + (size_t)la    * (size_t)D);
    const v4f* __restrict__ en = (const v4f*)(exemplars + (size_t)ln    * (size_t)D);

    const int D4 = D >> 2;
    float s1 = 0.f, s2 = 0.f, s3 = 0.f, s4 = 0.f;
    for (int k = lane; k < D4; k += 32) {
        v4f a = pa[k], n = pn[k], xa = ea[k], xn = en[k];
        s1 += fabsf(a.x - xa.x + EPS_F) + fabsf(a.y - xa.y + EPS_F)
            + fabsf(a.z - xa.z + EPS_F) + fabsf(a.w - xa.w + EPS_F);
        s2 += fabsf(n.x - xa.x + EPS_F) + fabsf(n.y - xa.y + EPS_F)
            + fabsf(n.z - xa.z + EPS_F) + fabsf(n.w - xa.w + EPS_F);
        s3 += fabsf(a.x - xn.x + EPS_F) + fabsf(a.y - xn.y + EPS_F)
            + fabsf(a.z - xn.z + EPS_F) + fabsf(a.w - xn.w + EPS_F);
        s4 += fabsf(n.x - xn.x + EPS_F) + fabsf(n.y - xn.y + EPS_F)
            + fabsf(n.z - xn.z + EPS_F) + fabsf(n.w - xn.w + EPS_F);
    }
#pragma unroll
    for (int off = 16; off >= 1; off >>= 1) {
        s1 += __shfl_xor(s1, off, 32);
        s2 += __shfl_xor(s2, off, 32);
        s3 += __shfl_xor(s3, off, 32);
        s4 += __shfl_xor(s4, off, 32);
    }
    if (lane == 0)
        hinge[gwave] = fmaxf(s1 - s2, 0.0f) + fmaxf(s4 - s3, 0.0f);
}

// ---------------------------------------------------------------------------
// Kernel 3: set flag=1 if any exemplar element is nonzero (or NaN).
// All writers store the same constant -> deterministic without atomics.
// ---------------------------------------------------------------------------
__global__ __launch_bounds__(256)
void norm_flag_kernel(const float* __restrict__ exemplars, long long n, int* flag) {
    const long long n4 = n >> 2;
    const v4f* __restrict__ p4 = (const v4f*)exemplars;
    bool nz = false;
    const long long stride = (long long)gridDim.x * blockDim.x;
    for (long long i = (long long)blockIdx.x * blockDim.x + threadIdx.x; i < n4; i += stride) {
        v4f v = p4[i];
        nz |= !(v.x == 0.0f) | !(v.y == 0.0f) | !(v.z == 0.0f) | !(v.w == 0.0f);
    }
    for (long long i = (n4 << 2) + (long long)blockIdx.x * blockDim.x + threadIdx.x;
         i < n; i += stride)
        nz |= !(exemplars[i] == 0.0f);
    if (nz) *flag = 1;
}

// ---------------------------------------------------------------------------
// Kernel 4: deterministic final reduction on the matrix pipe.
// Array length is pre-padded with zeros to a multiple of 64, so every
// iteration is one unconditional global_load_b64 + v_wmma_f32_16x16x4_f32:
//   D[m,n] += sum_k A[m,k]  with B = ones (every column holds the row sums).
// Any bijective placement of a 64-value chunk into A yields the same total,
// so lane L contributes values p[2L], p[2L+1]. Single wave, EXEC all-ones.
// ---------------------------------------------------------------------------
__device__ __forceinline__ float wmma_sum(const float* __restrict__ p, int n_pad, int lane) {
    v2f bones; bones.x = 1.0f; bones.y = 1.0f;
    v8f c = {};
    const v2f* __restrict__ p2 = (const v2f*)p;   // 64-value chunks = 32 v2f
    const int iters = n_pad >> 6;
    for (int it = 0; it < iters; ++it) {
        v2f a = p2[it * 32 + lane];               // unconditional b64 load
        c = __builtin_amdgcn_wmma_f32_16x16x4_f32(false, a, false, bones,
                                                  (short)0, c, false, false);
    }
    float t = c[0] + c[1] + c[2] + c[3] + c[4] + c[5] + c[6] + c[7]; // 8 rows, col n=lane%16
    float u = __shfl_xor(t, 16, 32);     // lane0 (M=0..7) + lane16 (M=8..15)
    return t + u;                        // valid on lane 0 (and 16)
}

__global__ __launch_bounds__(32)
void final_reduce_kernel(const float* __restrict__ row_loss, int nrows, int nrows_pad,
                         const float* __restrict__ hinge, int nh_pad,
                         const int* __restrict__ flag,
                         float* __restrict__ out) {
    const int lane = threadIdx.x;        // exactly one wave32, no divergence
    float sumS = wmma_sum(row_loss, nrows_pad, lane);
    float sumC = wmma_sum(hinge, nh_pad, lane);
    if (lane == 0) {
        float loss_softmax = sumS / (float)nrows;
        float loss_center  = sumC;
        float loss_total   = (*flag) ? (loss_softmax + LAMBDA_F * loss_center)
                                     : loss_softmax;
        out[0] = loss_total;
        out[1] = loss_softmax;
        out[2] = loss_center;
    }
}

// ---------------------------------------------------------------------------
// Host launch
// ---------------------------------------------------------------------------
extern "C" void kernel_launch(void* const* d_in, const int* in_sizes, int n_in,
                              void* d_out, int out_size, void* d_ws, size_t ws_size,
                              hipStream_t stream) {
    const float* anchor    = (const float*)d_in[0];
    const float* negative  = (const float*)d_in[1];
    const float* outputs   = (const float*)d_in[2];
    const int*   la        = (const int*)  d_in[3];
    const int*   ln        = (const int*)  d_in[4];
    const float* exemplars = (const float*)d_in[5];
    float* out = (float*)d_out;

    const int B = in_sizes[3];                 // 4096
    const int D = in_sizes[0] / B;             // 1024
    const int twoB = 2 * B;                    // 8192
    const int C = in_sizes[2] / twoB;          // 10000
    const long long nex = (long long)in_sizes[5];

    const int nrows_pad = round_up64(twoB);    // multiples of 64 for WMMA chunks
    const int nh_pad    = round_up64(B);

    float* row_loss = (float*)d_ws;            // [nrows_pad]
    float* hinge    = row_loss + nrows_pad;    // [nh_pad]
    int*   flag     = (int*)(hinge + nh_pad);  // [1]

    init_ws_kernel<<<1, 128, 0, stream>>>(row_loss, twoB, nrows_pad,
                                          hinge, B, nh_pad, flag);

    softmax_rows_kernel<<<twoB, 256, 0, stream>>>(outputs, la, ln, row_loss, B, C);

    const int center_blocks = (B * 32 + 255) / 256;
    center_kernel<<<center_blocks, 256, 0, stream>>>(anchor, negative, exemplars,
                                                     la, ln, hinge, B, D);

    norm_flag_kernel<<<1024, 256, 0, stream>>>(exemplars, nex, flag);

    final_reduce_kernel<<<1, 32, 0, stream>>>(row_loss, twoB, nrows_pad,
                                              hinge, nh_pad, flag, out);
}